// MultiLevelPhysicalEncoder_47201690583144
// MI455X (gfx1250) — compile-verified
//
#include <hip/hip_runtime.h>
#include <math.h>

#define BATCH 1024
#define LEN   16384
#define HALF  8192
#define EPSF  1e-8f
#define NT    256

typedef float v2f __attribute__((ext_vector_type(2)));
typedef float v8f __attribute__((ext_vector_type(8)));

// ---------------- deterministic block reductions (wave32) ----------------
__device__ inline float warpReduceSum(float v) {
#pragma unroll
  for (int o = 16; o > 0; o >>= 1) v += __shfl_xor(v, o, 32);
  return v;
}
__device__ inline float warpReduceMax(float v) {
#pragma unroll
  for (int o = 16; o > 0; o >>= 1) v = fmaxf(v, __shfl_xor(v, o, 32));
  return v;
}
__device__ inline float blockReduceSum(float v, float* rbuf) {
  v = warpReduceSum(v);
  int lane = threadIdx.x & 31, w = threadIdx.x >> 5;
  __syncthreads();
  if (lane == 0) rbuf[w] = v;
  __syncthreads();
  if (threadIdx.x == 0) {
    float s = 0.f;
    for (int i = 0; i < NT / 32; ++i) s += rbuf[i];
    rbuf[0] = s;
  }
  __syncthreads();
  float r = rbuf[0];
  __syncthreads();
  return r;
}
__device__ inline float blockReduceMax(float v, float* rbuf) {
  v = warpReduceMax(v);
  int lane = threadIdx.x & 31, w = threadIdx.x >> 5;
  __syncthreads();
  if (lane == 0) rbuf[w] = v;
  __syncthreads();
  if (threadIdx.x == 0) {
    float s = rbuf[0];
    for (int i = 1; i < NT / 32; ++i) s = fmaxf(s, rbuf[i]);
    rbuf[0] = s;
  }
  __syncthreads();
  float r = rbuf[0];
  __syncthreads();
  return r;
}

// ---------------- kernel 1: per-row FFT + 20 features (LDS-resident) ----------------
__global__ __launch_bounds__(NT) void feat_kernel(const float* __restrict__ sig,
                                                  float* __restrict__ feats) {
  __shared__ float sre[LEN];          // 64 KB
  __shared__ float sim[LEN];          // 64 KB  (320 KB/WGP LDS on CDNA5 makes this fit)
  __shared__ float rbuf[NT / 32];
  __shared__ float scan[NT];
  __shared__ int   sidx;

  const int tid = threadIdx.x;
  const int row = blockIdx.x;
  const float* x = sig + (size_t)row * LEN;

  for (int i = tid; i < LEN; i += NT) { sre[i] = x[i]; sim[i] = 0.f; }
  __syncthreads();

  // ---- time features (two passes over LDS copy of x) ----
  float s1 = 0.f, s2 = 0.f, pk = 0.f, sq = 0.f;
  for (int i = tid; i < LEN; i += NT) {
    float v = sre[i];
    s1 += v; s2 += v * v;
    pk = fmaxf(pk, fabsf(v));
    sq += sqrtf(fabsf(v) + EPSF);
  }
  s1 = blockReduceSum(s1, rbuf);
  s2 = blockReduceSum(s2, rbuf);
  pk = blockReduceMax(pk, rbuf);
  sq = blockReduceSum(sq, rbuf);
  float mean = s1 / (float)LEN;
  float var  = (s2 - (float)LEN * mean * mean) / (float)(LEN - 1);
  float stdd = sqrtf(fmaxf(var, 0.f));
  float rms  = sqrtf(s2 / (float)LEN);
  float msq  = sq / (float)LEN;
  float clearf = pk / (msq * msq + EPSF);

  float s3 = 0.f, s4 = 0.f;
  float izs = 1.f / (stdd + EPSF);
  for (int i = tid; i < LEN; i += NT) {
    float z = (sre[i] - mean) * izs;
    float z2 = z * z;
    s3 += z2 * z; s4 += z2 * z2;
  }
  s3 = blockReduceSum(s3, rbuf);
  s4 = blockReduceSum(s4, rbuf);
  float skew = s3 / (float)LEN;
  float kurt = s4 / (float)LEN - 3.0f;

  if (tid == 0) {
    float* f = feats + row * 20;
    f[0] = mean; f[1] = stdd; f[2] = var; f[3] = skew;
    f[4] = kurt; f[5] = rms;  f[6] = pk;  f[7] = clearf;
  }
  __syncthreads();

  // ---- in-place radix-2 DIT FFT in LDS ----
  for (int i = tid; i < LEN; i += NT) {
    int j = (int)(__brev((unsigned)i) >> 18);   // 14-bit reversal
    if (j > i) { float t = sre[i]; sre[i] = sre[j]; sre[j] = t; }  // sim is all-zero
  }
  __syncthreads();
  for (int s = 1; s <= 14; ++s) {
    const int half = 1 << (s - 1);
    const float w0 = -6.2831853071795864769f / (float)(1 << s);
    for (int t = tid; t < (LEN >> 1); t += NT) {
      int k  = t & (half - 1);
      int i0 = ((t >> (s - 1)) << s) + k;
      int i1 = i0 + half;
      float cw, sw;
      sincosf(w0 * (float)k, &sw, &cw);
      float ar = sre[i0], ai = sim[i0];
      float br = sre[i1], bi = sim[i1];
      float tr = br * cw - bi * sw;
      float ti = br * sw + bi * cw;
      sre[i0] = ar + tr; sim[i0] = ai + ti;
      sre[i1] = ar - tr; sim[i1] = ai - ti;
    }
    __syncthreads();
  }
  // magnitude of first HALF bins, overwrite sre
  for (int k = tid; k < HALF; k += NT) {
    float rr = sre[k], ii = sim[k];
    sre[k] = sqrtf(rr * rr + ii * ii);
  }
  __syncthreads();

  // ---- freq features pass 1: tp, msum, centroid numerator, band energies ----
  const float fscale = 0.5f / (float)(HALF - 1);
  const int bs = HALF / 5;              // 1638
  float tp = 0.f, ms = 0.f, fp = 0.f;
  float be0 = 0.f, be1 = 0.f, be2 = 0.f, be3 = 0.f, be4 = 0.f;
  for (int k = tid; k < HALF; k += NT) {
    float m = sre[k];
    float p = m * m;
    tp += p; ms += m;
    fp += fscale * (float)k * p;
    int j = k / bs; if (j > 4) j = 4;
    if      (j == 0) be0 += p;
    else if (j == 1) be1 += p;
    else if (j == 2) be2 += p;
    else if (j == 3) be3 += p;
    else             be4 += p;
  }
  tp  = blockReduceSum(tp, rbuf);
  ms  = blockReduceSum(ms, rbuf);
  fp  = blockReduceSum(fp, rbuf);
  be0 = blockReduceSum(be0, rbuf);
  be1 = blockReduceSum(be1, rbuf);
  be2 = blockReduceSum(be2, rbuf);
  be3 = blockReduceSum(be3, rbuf);
  be4 = blockReduceSum(be4, rbuf);
  float cent = fp / (tp + EPSF);

  // ---- pass 2: spectral moments + entropy ----
  float S2 = 0.f, S3 = 0.f, S4 = 0.f, EN = 0.f;
  float ims = 1.f / (ms + EPSF);
  for (int k = tid; k < HALF; k += NT) {
    float m = sre[k];
    float p = m * m;
    float d = fscale * (float)k - cent;
    float d2 = d * d;
    S2 += d2 * p; S3 += d2 * d * p; S4 += d2 * d2 * p;
    float nm = m * ims;
    EN += nm * logf(nm + EPSF);
  }
  S2 = blockReduceSum(S2, rbuf);
  S3 = blockReduceSum(S3, rbuf);
  S4 = blockReduceSum(S4, rbuf);
  EN = blockReduceSum(EN, rbuf);
  float svar  = S2 / (tp + EPSF);
  float sskew = S3 / (tp * powf(svar, 1.5f) + EPSF);
  float skurt = S4 / (tp * svar * svar + EPSF) - 3.0f;
  float ent = -EN;

  // ---- rolloff: first k with inclusive cumsum(mag) >= 0.85*msum ----
  if (tid == 0) sidx = HALF - 1;
  const int chunk = HALF / NT;          // 32 contiguous bins per thread
  int base = tid * chunk;
  float ls = 0.f;
  for (int e = 0; e < chunk; ++e) ls += sre[base + e];
  scan[tid] = ls;
  __syncthreads();
  for (int off = 1; off < NT; off <<= 1) {
    float t = (tid >= off) ? scan[tid - off] : 0.f;
    __syncthreads();
    scan[tid] += t;
    __syncthreads();
  }
  float run = scan[tid] - ls;           // exclusive prefix for this chunk
  float thr = 0.85f * ms;
  int found = -1;
  for (int e = 0; e < chunk; ++e) {
    run += sre[base + e];
    if (run >= thr) { found = base + e; break; }
  }
  if (found >= 0) atomicMin(&sidx, found);   // integer min: order-independent
  __syncthreads();
  float roll = fscale * (float)sidx;

  if (tid == 0) {
    float* f = feats + row * 20;
    f[8]  = cent; f[9] = svar; f[10] = sskew; f[11] = skurt;
    f[12] = tp;   f[13] = ent; f[14] = roll;
    float bsum = be0 + be1 + be2 + be3 + be4;
    float ib = 1.f / (bsum + EPSF);
    f[15] = be0 * ib; f[16] = be1 * ib; f[17] = be2 * ib;
    f[18] = be3 * ib; f[19] = be4 * ib;
  }
}

// ---------------- kernel 2: three tiny MLPs, one thread per row ----------------
__global__ __launch_bounds__(NT) void small_mlps(
    const float* __restrict__ feats,
    const float* __restrict__ tW1, const float* __restrict__ tb1,
    const float* __restrict__ tW2, const float* __restrict__ tb2,
    const float* __restrict__ fW1, const float* __restrict__ fb1,
    const float* __restrict__ fW2, const float* __restrict__ fb2,
    const float* __restrict__ gW1, const float* __restrict__ gb1,
    const float* __restrict__ gW2, const float* __restrict__ gb2,
    float* __restrict__ comb) {
  int r = blockIdx.x * blockDim.x + threadIdx.x;
  if (r >= BATCH) return;
  const float* in = feats + r * 20;
  float out[48];
  {   // time: 8 -> 32 -> 16, relu/relu
    float h[32];
    for (int o = 0; o < 32; ++o) {
      float a = tb1[o];
      for (int i = 0; i < 8; ++i) a += in[i] * tW1[i * 32 + o];
      h[o] = fmaxf(a, 0.f);
    }
    for (int o = 0; o < 16; ++o) {
      float a = tb2[o];
      for (int i = 0; i < 32; ++i) a += h[i] * tW2[i * 16 + o];
      out[o] = fmaxf(a, 0.f);
    }
  }
  {   // freq: 7 -> 28 -> 16
    float h[28];
    const float* xi = in + 8;
    for (int o = 0; o < 28; ++o) {
      float a = fb1[o];
      for (int i = 0; i < 7; ++i) a += xi[i] * fW1[i * 28 + o];
      h[o] = fmaxf(a, 0.f);
    }
    for (int o = 0; o < 16; ++o) {
      float a = fb2[o];
      for (int i = 0; i < 28; ++i) a += h[i] * fW2[i * 16 + o];
      out[16 + o] = fmaxf(a, 0.f);
    }
  }
  {   // band: 5 -> 20 -> 16
    float h[20];
    const float* xi = in + 15;
    for (int o = 0; o < 20; ++o) {
      float a = gb1[o];
      for (int i = 0; i < 5; ++i) a += xi[i] * gW1[i * 20 + o];
      h[o] = fmaxf(a, 0.f);
    }
    for (int o = 0; o < 16; ++o) {
      float a = gb2[o];
      for (int i = 0; i < 20; ++i) a += h[i] * gW2[i * 16 + o];
      out[32 + o] = fmaxf(a, 0.f);
    }
  }
  float* c = comb + r * 48;
  for (int i = 0; i < 48; ++i) c[i] = out[i];
}

// ---------------- kernel 3: u-MLP (48->128 relu, 128->64 tanh) via f32 WMMA ----------------
// Features can reach ~1e8 (total spectral power), so fp16 inputs would overflow;
// V_WMMA_F32_16X16X4_F32 keeps full fp32 A/B/C precision.
// Fragment layouts per ISA 7.12.2:
//   A 16x4:  lanes 0-15 -> M=lane, K={0,1}; lanes 16-31 -> M=lane-16, K={2,3}
//   B 4x16:  lanes 0-15 -> N=lane, K={0,1}; lanes 16-31 -> N=lane-16, K={2,3}
//   C/D 16x16: VGPR i: M=i (lanes 0-15) / M=i+8 (lanes 16-31), N=lane&15
__global__ __launch_bounds__(NT) void umlp_wmma(
    const float* __restrict__ comb,
    const float* __restrict__ uW1, const float* __restrict__ ub1,
    const float* __restrict__ uW2, const float* __restrict__ ub2,
    float* __restrict__ code) {
  __shared__ float sA[16 * 48];
  __shared__ float sW1[48 * 128];
  __shared__ float sH[16 * 128];
  __shared__ float sW2[128 * 64];

  const int tid  = threadIdx.x;
  const int row0 = blockIdx.x * 16;

  for (int i = tid; i < 16 * 48;  i += NT) sA[i]  = comb[row0 * 48 + i];
  for (int i = tid; i < 48 * 128; i += NT) sW1[i] = uW1[i];
  for (int i = tid; i < 128 * 64; i += NT) sW2[i] = uW2[i];
  __syncthreads();

  const int wave = tid >> 5;
  const int lane = tid & 31;
  const int rl   = lane & 15;
  const int hi   = lane >> 4;

  // layer 1: each wave owns a 16x16 output tile, 8 waves cover N=128
  {
    const int nb = wave * 16;
    v8f acc = {};
    for (int kk = 0; kk < 48; kk += 4) {
      int k0 = kk + 2 * hi;
      v2f a, b;
      a.x = sA[rl * 48 + k0];
      a.y = sA[rl * 48 + k0 + 1];
      b.x = sW1[k0 * 128 + nb + rl];
      b.y = sW1[(k0 + 1) * 128 + nb + rl];
      acc = __builtin_amdgcn_wmma_f32_16x16x4_f32(false, a, false, b,
                                                  (short)0, acc, false, false);
    }
#pragma unroll
    for (int i = 0; i < 8; ++i) {
      int m = i + 8 * hi;
      int n = nb + rl;
      sH[m * 128 + n] = fmaxf(acc[i] + ub1[n], 0.f);
    }
  }
  __syncthreads();

  // layer 2: waves 0..3 cover N=64 (wave-uniform branch keeps EXEC all-ones)
  if (wave < 4) {
    const int nb = wave * 16;
    v8f acc = {};
    for (int kk = 0; kk < 128; kk += 4) {
      int k0 = kk + 2 * hi;
      v2f a, b;
      a.x = sH[rl * 128 + k0];
      a.y = sH[rl * 128 + k0 + 1];
      b.x = sW2[k0 * 64 + nb + rl];
      b.y = sW2[(k0 + 1) * 64 + nb + rl];
      acc = __builtin_amdgcn_wmma_f32_16x16x4_f32(false, a, false, b,
                                                  (short)0, acc, false, false);
    }
#pragma unroll
    for (int i = 0; i < 8; ++i) {
      int m = i + 8 * hi;
      int n = nb + rl;
      code[(size_t)(row0 + m) * 64 + n] = tanhf(acc[i] + ub2[n]);
    }
  }
}

// ---------------- kernel 4: batch-norm across batch dim ----------------
__global__ __launch_bounds__(NT) void bn_kernel(const float* __restrict__ code,
                                                const float* __restrict__ gamma,
                                                const float* __restrict__ beta,
                                                float* __restrict__ out) {
  __shared__ float rbuf[NT / 32];
  const int c = blockIdx.x;
  const int tid = threadIdx.x;
  float s = 0.f, s2 = 0.f;
  for (int r = tid; r < BATCH; r += NT) {
    float v = code[r * 64 + c];
    s += v; s2 += v * v;
  }
  s  = blockReduceSum(s, rbuf);
  s2 = blockReduceSum(s2, rbuf);
  float m   = s / (float)BATCH;
  float var = s2 / (float)BATCH - m * m;
  float inv = 1.0f / sqrtf(fmaxf(var, 0.f) + 1e-5f);
  float g = gamma[c], bb = beta[c];
  for (int r = tid; r < BATCH; r += NT) {
    out[r * 64 + c] = (code[r * 64 + c] - m) * inv * g + bb;
  }
}

extern "C" void kernel_launch(void* const* d_in, const int* in_sizes, int n_in,
                              void* d_out, int out_size, void* d_ws, size_t ws_size,
                              hipStream_t stream) {
  const float* sig  = (const float*)d_in[0];
  const float* tW1  = (const float*)d_in[1];
  const float* tb1  = (const float*)d_in[2];
  const float* tW2  = (const float*)d_in[3];
  const float* tb2  = (const float*)d_in[4];
  const float* fW1  = (const float*)d_in[5];
  const float* fb1  = (const float*)d_in[6];
  const float* fW2  = (const float*)d_in[7];
  const float* fb2  = (const float*)d_in[8];
  const float* gW1  = (const float*)d_in[9];
  const float* gb1  = (const float*)d_in[10];
  const float* gW2  = (const float*)d_in[11];
  const float* gb2  = (const float*)d_in[12];
  const float* uW1  = (const float*)d_in[13];
  const float* ub1  = (const float*)d_in[14];
  const float* uW2  = (const float*)d_in[15];
  const float* ub2  = (const float*)d_in[16];
  const float* gam  = (const float*)d_in[17];
  const float* bet  = (const float*)d_in[18];

  float* feats = (float*)d_ws;               // 1024 * 20
  float* comb  = feats + BATCH * 20;         // 1024 * 48
  float* code  = comb  + BATCH * 48;         // 1024 * 64

  feat_kernel<<<BATCH, NT, 0, stream>>>(sig, feats);
  small_mlps<<<(BATCH + NT - 1) / NT, NT, 0, stream>>>(
      feats, tW1, tb1, tW2, tb2, fW1, fb1, fW2, fb2, gW1, gb1, gW2, gb2, comb);
  umlp_wmma<<<BATCH / 16, NT, 0, stream>>>(comb, uW1, ub1, uW2, ub2, code);
  bn_kernel<<<64, NT, 0, stream>>>(code, gam, bet, (float*)d_out);
}